// Encoder_Postnet_combine_62904091017716
// MI455X (gfx1250) — compile-verified
//
#include <hip/hip_runtime.h>

#define B_ 16
#define T_ 4096
#define P_ 512
#define D_ 512
#define S_ 100

typedef __attribute__((ext_vector_type(2))) float v2f;
typedef __attribute__((ext_vector_type(4))) float v4f;
typedef __attribute__((ext_vector_type(8))) float v8f;

// ---------------------------------------------------------------------------
// 1) Per-batch serial alignment scan (faithful to the lax.scan semantics).
//    B=16 scans of T=4096 steps: negligible vs. the 134MB output pass.
// ---------------------------------------------------------------------------
__global__ void align_scan_kernel(const int* __restrict__ align,
                                  const int* __restrict__ text,
                                  int* __restrict__ idx) {
  if (threadIdx.x != 0) return;
  const int b = blockIdx.x;
  const int* al = align + (size_t)b * T_;
  const int* tx = text + (size_t)b * P_;
  int ind = 0;
  int before = tx[0];
  idx[(size_t)b * T_] = 0;
  for (int t = 1; t < T_; ++t) {
    const int a = al[t];
    if (a != before) {
      ind = ind + 1;
      if (ind > P_ - 1) ind = P_ - 1;
      before = tx[ind];
    }
    idx[(size_t)b * T_ + t] = ind;
  }
}

// ---------------------------------------------------------------------------
// 2) Sinusoidal positional-encoding table pe[T_, D_].
// ---------------------------------------------------------------------------
__global__ void pe_kernel(float* __restrict__ pe) {
  const int t = blockIdx.x;          // 0..T_-1
  const int i = threadIdx.x;         // pair index 0..D_/2-1
  const float c = -9.210340371976184f / (float)D_;   // -ln(10000)/D
  const float div = __expf((float)(2 * i) * c);
  float s, co;
  __sincosf((float)t * div, &s, &co);
  pe[(size_t)t * D_ + 2 * i]     = s;
  pe[(size_t)t * D_ + 2 * i + 1] = co;
}

// ---------------------------------------------------------------------------
// 3) f32 WMMA GEMM:  C[M,N] = A[M,K] @ Bm[K,N]  (+ add[M,N] if add != null)
//    One wave (32 threads) per block computes a 32x32 C tile via 2x2 grid of
//    16x16 tiles, K stepped by 4 with V_WMMA_F32_16X16X4_F32.
//    N, K multiples of 32/4; M handled by clamped reads + guarded writes.
// ---------------------------------------------------------------------------
__global__ void gemm_f32_wmma(const float* __restrict__ A,
                              const float* __restrict__ Bm,
                              const float* __restrict__ add,
                              float* __restrict__ C,
                              int M, int N, int K) {
  const int lane = threadIdx.x & 31;
  const int half = lane >> 4;     // 0: lanes 0-15, 1: lanes 16-31
  const int l16  = lane & 15;
  const int tileN = blockIdx.x * 32;
  const int tileM = blockIdx.y * 32;

  v8f acc[2][2] = {};

  int rowA[2];
  rowA[0] = tileM + l16;
  rowA[1] = tileM + 16 + l16;
  if (rowA[0] >= M) rowA[0] = M - 1;   // clamp reads; writes guarded below
  if (rowA[1] >= M) rowA[1] = M - 1;

  for (int k0 = 0; k0 < K; k0 += 4) {
    v2f a[2], b[2];
    // A frag: lane(l16,half) holds A[row][k0+2*half+v], v=0,1 (ISA 16x4 f32 layout)
#pragma unroll
    for (int i = 0; i < 2; ++i) {
      const float* ap = A + (size_t)rowA[i] * K + (k0 + 2 * half);
      a[i].x = ap[0];
      a[i].y = ap[1];
    }
    // B frag: lane holds B[k0+2*half+v][col], v=0,1
#pragma unroll
    for (int j = 0; j < 2; ++j) {
      const float* bp = Bm + (size_t)(k0 + 2 * half) * N + (tileN + j * 16 + l16);
      b[j].x = bp[0];
      b[j].y = bp[N];
    }
#pragma unroll
    for (int i = 0; i < 2; ++i)
#pragma unroll
      for (int j = 0; j < 2; ++j)
        acc[i][j] = __builtin_amdgcn_wmma_f32_16x16x4_f32(
            /*neg_a=*/false, a[i], /*neg_b=*/false, b[j],
            /*c_mod=*/(short)0, acc[i][j],
            /*reuse_a=*/false, /*reuse_b=*/false);
  }

  // C/D layout: VGPR v -> M = v (lanes 0-15) / v+8 (lanes 16-31), N = l16
#pragma unroll
  for (int i = 0; i < 2; ++i) {
#pragma unroll
    for (int j = 0; j < 2; ++j) {
      const int col = tileN + j * 16 + l16;
#pragma unroll
      for (int v = 0; v < 8; ++v) {
        const int row = tileM + i * 16 + v + 8 * half;
        if (row < M) {
          float val = acc[i][j][v];
          if (add) val += add[(size_t)row * N + col];
          C[(size_t)row * N + col] = val;
        }
      }
    }
  }
}

// ---------------------------------------------------------------------------
// 4) Small projection vectors through w1 = w_out[0:D,:]:
//    vecs[0] = w_pitch @ w1          (pitch direction)
//    vecs[1] = (b_pitch+b_pos)@w1 + b_out   (constant)
//    vecs[2] = emb_beats[0] @ w1
//    vecs[3] = emb_beats[1] @ w1
// ---------------------------------------------------------------------------
__global__ void small_vecs_kernel(const float* __restrict__ w_out,
                                  const float* __restrict__ w_pitch,
                                  const float* __restrict__ b_pitch,
                                  const float* __restrict__ b_pos,
                                  const float* __restrict__ emb_beats,
                                  const float* __restrict__ b_out,
                                  float* __restrict__ vecs) {
  const int d = blockIdx.x * blockDim.x + threadIdx.x;
  if (d >= D_) return;
  float pd = 0.f, cv = 0.f, bt0 = 0.f, bt1 = 0.f;
  for (int k = 0; k < D_; ++k) {
    const float w = w_out[(size_t)k * D_ + d];
    pd  += w_pitch[k] * w;
    cv  += (b_pitch[k] + b_pos[k]) * w;
    bt0 += emb_beats[k] * w;
    bt1 += emb_beats[D_ + k] * w;
  }
  vecs[d]          = pd;
  vecs[D_ + d]     = cv + b_out[d];
  vecs[2 * D_ + d] = bt0;
  vecs[3 * D_ + d] = bt1;
}

// ---------------------------------------------------------------------------
// 5) Fused streaming combine + leaky ReLU (bandwidth-bound pass).
//    out[b,t,:] = Aw[b*P+idx[b,t],:] + PEt[t,:] + pitch*vecs[0] + vecs[2+beat]
//                 + ST[singer,:] + vecs[1], then leaky_relu(0.01).
//    The 134MB output is written with non-temporal hints so the L2 keeps the
//    heavily reused Aw/PEt/ST read-set resident (8-16x row reuse).
// ---------------------------------------------------------------------------
__device__ __forceinline__ float lrelu(float x) { return x >= 0.f ? x : 0.01f * x; }

__global__ void fuse_kernel(const float* __restrict__ Aw,
                            const float* __restrict__ PEt,
                            const float* __restrict__ ST,
                            const float* __restrict__ vecs,
                            const int* __restrict__ idx,
                            const float* __restrict__ pitch,
                            const int* __restrict__ beats,
                            const int* __restrict__ sv,
                            float* __restrict__ out) {
  const int t = blockIdx.x;
  const int b = blockIdx.y;
  const int bt = b * T_ + t;
  const int row = b * P_ + idx[bt];
  const float p = pitch[bt];
  const int e = beats[bt] & 1;
  int s = sv[bt];
  s = s < 0 ? 0 : (s >= S_ ? S_ - 1 : s);
  const int d4 = threadIdx.x;  // 128 threads * 4 floats = 512 floats

  const v4f a  = ((const v4f*)(Aw  + (size_t)row * D_))[d4];
  const v4f pq = ((const v4f*)(PEt + (size_t)t   * D_))[d4];
  const v4f st = ((const v4f*)(ST  + (size_t)s   * D_))[d4];
  const v4f pd = ((const v4f*)(vecs))[d4];
  const v4f cv = ((const v4f*)(vecs + D_))[d4];
  const v4f bb = ((const v4f*)(vecs + (size_t)(2 + e) * D_))[d4];

  v4f r;
  r.x = lrelu(a.x + pq.x + p * pd.x + bb.x + st.x + cv.x);
  r.y = lrelu(a.y + pq.y + p * pd.y + bb.y + st.y + cv.y);
  r.z = lrelu(a.z + pq.z + p * pd.z + bb.z + st.z + cv.z);
  r.w = lrelu(a.w + pq.w + p * pd.w + bb.w + st.w + cv.w);

  // Non-temporal streaming store: output is never re-read.
  v4f* dst = (v4f*)(out + (size_t)bt * D_) + d4;
  __builtin_nontemporal_store(r, dst);
}

// ---------------------------------------------------------------------------
extern "C" void kernel_launch(void* const* d_in, const int* in_sizes, int n_in,
                              void* d_out, int out_size, void* d_ws, size_t ws_size,
                              hipStream_t stream) {
  const float* encoder_out = (const float*)d_in[0];   // [B,P,D]
  const int*   align_phone = (const int*)  d_in[1];   // [B,T]
  const int*   text_phone  = (const int*)  d_in[2];   // [B,P]
  const float* pitch       = (const float*)d_in[3];   // [B,T,1]
  const int*   beats       = (const int*)  d_in[4];   // [B,T,1]
  const int*   singer_vec  = (const int*)  d_in[5];   // [B,T,1]
  const float* w_pitch     = (const float*)d_in[6];   // [1,D]
  const float* b_pitch     = (const float*)d_in[7];   // [D]
  const float* w_pos       = (const float*)d_in[8];   // [D,D]
  const float* b_pos       = (const float*)d_in[9];   // [D]
  const float* emb_beats   = (const float*)d_in[10];  // [2,D]
  const float* emb_singer  = (const float*)d_in[11];  // [S,D]
  const float* w_out       = (const float*)d_in[12];  // [2D,D]  (w1 = rows 0..D-1, w2 = rows D..2D-1)
  const float* b_out       = (const float*)d_in[13];  // [D]
  float* out = (float*)d_out;

  // --- carve workspace ---
  char* ws = (char*)d_ws;
  size_t off = 0;
  auto carve = [&](size_t bytes) -> void* {
    void* p = ws + off;
    off = (off + bytes + 255) & ~(size_t)255;
    return p;
  };
  int*   idx  = (int*)  carve((size_t)B_ * T_ * 4);        // [B,T]
  float* Wp1  = (float*)carve((size_t)D_ * D_ * 4);        // w_pos @ w1
  float* Wc   = (float*)carve((size_t)D_ * D_ * 4);        // w1 + w_pos @ w1
  float* Aw   = (float*)carve((size_t)B_ * P_ * D_ * 4);   // encoder_out @ Wc
  float* PE   = (float*)carve((size_t)T_ * D_ * 4);        // pe table
  float* PEt  = (float*)carve((size_t)T_ * D_ * 4);        // pe @ Wp1
  float* ST   = (float*)carve((size_t)128 * D_ * 4);       // emb_singer @ w2 (padded)
  float* vecs = (float*)carve((size_t)4 * D_ * 4);         // small projection vectors

  const float* w1 = w_out;
  const float* w2 = w_out + (size_t)D_ * D_;

  // 1) alignment indices
  align_scan_kernel<<<B_, 32, 0, stream>>>(align_phone, text_phone, idx);

  // 2) positional encoding table
  pe_kernel<<<T_, D_ / 2, 0, stream>>>(PE);

  // 3) precompute GEMMs (f32 WMMA)
  gemm_f32_wmma<<<dim3(D_ / 32, D_ / 32), 32, 0, stream>>>(w_pos, w1, w1,     Wc,  D_,      D_, D_);
  gemm_f32_wmma<<<dim3(D_ / 32, D_ / 32), 32, 0, stream>>>(w_pos, w1, nullptr, Wp1, D_,      D_, D_);
  gemm_f32_wmma<<<dim3(D_ / 32, (B_ * P_) / 32), 32, 0, stream>>>(encoder_out, Wc, nullptr, Aw,  B_ * P_, D_, D_);
  gemm_f32_wmma<<<dim3(D_ / 32, T_ / 32), 32, 0, stream>>>(PE, Wp1, nullptr, PEt, T_,      D_, D_);
  gemm_f32_wmma<<<dim3(D_ / 32, (S_ + 31) / 32), 32, 0, stream>>>(emb_singer, w2, nullptr, ST,  S_,      D_, D_);

  // 4) small vectors
  small_vecs_kernel<<<dim3((D_ + 255) / 256), 256, 0, stream>>>(
      w1, w_pitch, b_pitch, b_pos, emb_beats, b_out, vecs);

  // 5) fused bandwidth-bound combine (134 MB NT store)
  fuse_kernel<<<dim3(T_, B_), D_ / 4, 0, stream>>>(
      Aw, PEt, ST, vecs, idx, pitch, beats, singer_vec, out);
}